// InstanceLoss_22196390986042
// MI455X (gfx1250) — compile-verified
//
#include <hip/hip_runtime.h>
#include <math.h>

// ---- problem constants (from reference setup_inputs) ----
#define BATCH 4
#define CCH   64
#define HWPX  (512 * 512)      // 262144 pixels per image
#define KINST 16
#define NCLS  11
#define EPSR  1e-8f

// ---- tiling ----
#define GCH   128              // chunks per image
#define CHUNK 2048             // pixels per chunk; GCH*CHUNK == HWPX
#define PART  (2 * CCH * KINST + KINST)   // 2064 floats per partial: A1[64x16], Y2[64x16], cnt[16]

typedef float v2f __attribute__((ext_vector_type(2)));
typedef float v8f __attribute__((ext_vector_type(8)));

// ---------------------------------------------------------------------------
// Phase 1: per (image, pixel-chunk) masked reductions via V_WMMA_F32_16X16X4_F32
//   A1[c,k] = sum_p X1[c,p] * M[k,p]
//   Y2[c,k] = sum_p (X2[c,p] / max(||X2[:,p]||, eps)) * M[k,p]
//   cnt[k]  = sum_p M[k,p]          (ones-row WMMA; all rows of D identical)
// Block = 128 threads = 4 waves; wave w owns channel rows [16w, 16w+16).
// All WMMAs execute unconditionally -> EXEC is all-1s for every WMMA (ISA req).
// ---------------------------------------------------------------------------
__global__ __launch_bounds__(128) void il_phase1(const float* __restrict__ x1,
                                                 const float* __restrict__ x2,
                                                 const int*   __restrict__ masks,
                                                 float*       __restrict__ part) {
    __shared__ float          s_rn[CHUNK];     // 1/max(||x2 pixel||, eps)
    __shared__ unsigned short s_bits[CHUNK];   // 16 instance-mask bits per pixel

    const int blk = blockIdx.x;
    const int b   = blk / GCH;
    const int g   = blk % GCH;
    const int p0  = g * CHUNK;
    const int tid = threadIdx.x;
    const size_t imgOff = (size_t)b * CCH * HWPX;

    // ---- Stage A: per-pixel inverse norm of view2 + packed mask bits ----
    for (int i = tid; i < CHUNK; i += 128) {
        const int p = p0 + i;
        const float* px2 = x2 + imgOff + p;
        float ss = 0.f;
        #pragma unroll 8
        for (int c = 0; c < CCH; ++c) {
            float v = px2[(size_t)c * HWPX];   // coalesced across threads
            ss = fmaf(v, v, ss);
        }
        s_rn[i] = 1.f / fmaxf(sqrtf(ss), EPSR);

        const int* pm = masks + (size_t)b * KINST * HWPX + p;
        unsigned int bits = 0u;
        #pragma unroll
        for (int k = 0; k < KINST; ++k)
            bits |= (pm[(size_t)k * HWPX] != 0 ? 1u : 0u) << k;
        s_bits[i] = (unsigned short)bits;
    }
    __syncthreads();

    // ---- Stage B: WMMA accumulation over the chunk ----
    const int wv   = tid >> 5;        // wave id 0..3 -> channel tile
    const int lane = tid & 31;
    const int r    = lane & 15;       // A: row-in-tile; B: instance column
    const int hi   = lane >> 4;       // lane half selects pixel pair {0,1} vs {2,3}
    const int c    = wv * 16 + r;

    const float* rx1 = x1 + imgOff + (size_t)c * HWPX + p0;
    const float* rx2 = x2 + imgOff + (size_t)c * HWPX + p0;

    v8f acc1 = {};   // X1 @ M^T   (channel tile x 16 instances)
    v8f accY = {};   // X2n @ M^T
    v8f accC = {};   // ones @ M^T -> per-instance pixel counts
    const v2f ones = {1.f, 1.f};

    // 32 pixels per group = one 128B cache line per row; prefetch 128 px (four lines)
    // ahead at WGP scope (locality 3 -> fills all cache levels, incl. L0).
    for (int p = 0; p < CHUNK; p += 32) {
        const int qp = p + 2 * hi + 128;
        __builtin_prefetch(rx1 + qp, 0, 3);
        __builtin_prefetch(rx2 + qp, 0, 3);
        #pragma unroll
        for (int pp = 0; pp < 32; pp += 4) {
            const int q = p + pp + 2 * hi;     // this lane's pixel pair
            // A operands (16x4 f32): v0/v1 = this lane's two K(pixel) slots
            v2f a1 = { rx1[q], rx1[q + 1] };
            v2f a2 = { rx2[q] * s_rn[q], rx2[q + 1] * s_rn[q + 1] };
            // B operand (4x16 f32): row = pixel, col = instance (lane%16)
            const unsigned int bq0 = s_bits[q], bq1 = s_bits[q + 1];
            v2f bm = { (float)((bq0 >> r) & 1u), (float)((bq1 >> r) & 1u) };

            acc1 = __builtin_amdgcn_wmma_f32_16x16x4_f32(false, a1, false, bm,
                                                         (short)0, acc1, false, false);
            accY = __builtin_amdgcn_wmma_f32_16x16x4_f32(false, a2, false, bm,
                                                         (short)0, accY, false, false);
            accC = __builtin_amdgcn_wmma_f32_16x16x4_f32(false, ones, false, bm,
                                                         (short)0, accC, false, false);
        }
    }

    // ---- Store partials (C/D layout: vgpr v -> row v (lanes 0-15) / v+8 (lanes 16-31)) ----
    float* out = part + (size_t)blk * PART;
    #pragma unroll
    for (int v = 0; v < 8; ++v) {
        const int row = wv * 16 + v + hi * 8;       // channel
        out[row * KINST + r]                   = acc1[v];
        out[CCH * KINST + row * KINST + r]     = accY[v];
    }
    if (wv == 0 && hi == 0)                         // row 0 of count tile (stores only)
        out[2 * CCH * KINST + r] = accC[0];
}

// ---------------------------------------------------------------------------
// Phase 2: deterministic reduction of GCH chunk-partials per image.
// ---------------------------------------------------------------------------
__global__ void il_phase2(const float* __restrict__ part, float* __restrict__ red) {
    const int idx = blockIdx.x * blockDim.x + threadIdx.x;
    if (idx >= BATCH * PART) return;
    const int b = idx / PART;
    const int e = idx % PART;
    const float* p = part + (size_t)b * GCH * PART + e;
    float s = 0.f;
    for (int g = 0; g < GCH; ++g) s += p[(size_t)g * PART];
    red[idx] = s;
}

// ---------------------------------------------------------------------------
// Phase 3: tiny epilogue. sim[i,j] = Mn[:,i] . Y2[:,j] / cnt_j, then label
// aggregation across all 4 images; writes the 23 output floats.
// ---------------------------------------------------------------------------
__global__ __launch_bounds__(256) void il_phase3(const float* __restrict__ red,
                                                 const int*   __restrict__ labels,
                                                 float*       __restrict__ out) {
    __shared__ float sInvC[KINST];           // 1/cnt_s
    __shared__ float sInvNC[KINST];          // (1/cnt_s) * 1/max(||mean||, eps)
    __shared__ float sSim[KINST][KINST];
    __shared__ float aInst[NCLS], aCls[NCLS], aCnt[NCLS], aNeg;

    const int tid = threadIdx.x;
    if (tid < NCLS) { aInst[tid] = 0.f; aCls[tid] = 0.f; aCnt[tid] = 0.f; }
    if (tid == 0) aNeg = 0.f;
    __syncthreads();

    for (int b = 0; b < BATCH; ++b) {
        const float* A1 = red + (size_t)b * PART;
        const float* Y2 = A1 + CCH * KINST;
        const float* CT = A1 + 2 * CCH * KINST;

        if (tid < KINST) {
            const float cs  = fmaxf(CT[tid], 1.f);
            const float inv = 1.f / cs;
            float ss = 0.f;
            for (int c = 0; c < CCH; ++c) {
                const float m = A1[c * KINST + tid] * inv;   // mean[c, tid]
                ss = fmaf(m, m, ss);
            }
            sInvC[tid]  = inv;
            sInvNC[tid] = inv / fmaxf(sqrtf(ss), EPSR);
        }
        __syncthreads();

        {   // 256 threads = full 16x16 sim matrix
            const int i = tid >> 4, j = tid & 15;
            float s = 0.f;
            for (int c = 0; c < CCH; ++c)
                s = fmaf(A1[c * KINST + i], Y2[c * KINST + j], s);
            sSim[i][j] = s * sInvNC[i] * sInvC[j];
        }
        __syncthreads();

        if (tid == 0) {
            const int* lab = labels + b * KINST;
            float negS = 0.f, negC = 0.f;
            for (int i = 0; i < KINST; ++i) {
                const int li = lab[i];
                aInst[li] += sSim[i][i];
                aCnt[li]  += 1.f;
                for (int j = 0; j < KINST; ++j) {
                    if (j == i) continue;
                    if (lab[j] == li) aCls[li] += sSim[i][j];
                    else { negS += sSim[i][j]; negC += 1.f; }
                }
            }
            aNeg += negS / fmaxf(negC, 1.f);   // per-image normalization, then summed
        }
        __syncthreads();
    }

    if (tid < NCLS) {
        const float cnt = aCnt[tid];
        out[tid]        = (cnt > 0.f) ? aInst[tid] / fmaxf(cnt, 1.f) : 0.f;
        out[NCLS + tid] = (cnt > 1.f) ? aCls[tid] / fmaxf(cnt * (cnt - 1.f), 1.f) : 0.f;
    }
    if (tid == 0) out[2 * NCLS] = aNeg;
}

// ---------------------------------------------------------------------------
extern "C" void kernel_launch(void* const* d_in, const int* in_sizes, int n_in,
                              void* d_out, int out_size, void* d_ws, size_t ws_size,
                              hipStream_t stream) {
    const float* x1     = (const float*)d_in[0];   // views_1 [4,64,512,512] f32
    const float* x2     = (const float*)d_in[1];   // views_2 [4,64,512,512] f32
    const int*   masks  = (const int*)d_in[2];     // masks   [4,16,512,512] i32
    const int*   labels = (const int*)d_in[3];     // labels  [4,16] i32
    float*       out    = (float*)d_out;           // 11 + 11 + 1 = 23 f32

    // workspace: partials (BATCH*GCH*PART floats ~= 4.2 MB) + reduced (BATCH*PART)
    float* part = (float*)d_ws;
    float* red  = part + (size_t)BATCH * GCH * PART;

    il_phase1<<<BATCH * GCH, 128, 0, stream>>>(x1, x2, masks, part);

    const int n2 = BATCH * PART;
    il_phase2<<<(n2 + 255) / 256, 256, 0, stream>>>(part, red);

    il_phase3<<<1, 256, 0, stream>>>(red, labels, out);
}